// TAG_nofc_L3_70574902608031
// MI455X (gfx1250) — compile-verified
//
#include <hip/hip_runtime.h>
#include <math.h>

#define NN 100000
#define EE 3200000

typedef float v2f __attribute__((ext_vector_type(2)));
typedef float v8f __attribute__((ext_vector_type(8)));

// ---------------- utility kernels ----------------

__global__ __launch_bounds__(256) void zero_kernel(float* __restrict__ p, long long n) {
  long long i = (long long)blockIdx.x * 256 + threadIdx.x;
  if (i < n) p[i] = 0.0f;
}

__global__ __launch_bounds__(256) void relu_kernel(float* __restrict__ p, long long n) {
  long long i = (long long)blockIdx.x * 256 + threadIdx.x;
  if (i < n) p[i] = fmaxf(p[i], 0.0f);
}

// ---------------- gcn_norm ----------------

__global__ __launch_bounds__(256) void deg_kernel(const int* __restrict__ dst,
                                                  const float* __restrict__ ew,
                                                  float* __restrict__ deg) {
  int e = blockIdx.x * 256 + threadIdx.x;
  if (e < EE) atomicAdd(&deg[dst[e]], ew[e]);
}

__global__ __launch_bounds__(256) void invsqrt_kernel(float* __restrict__ deg) {
  int i = blockIdx.x * 256 + threadIdx.x;
  if (i < NN) {
    float d = deg[i];
    deg[i] = (d > 0.0f) ? rsqrtf(d) : 0.0f;
  }
}

__global__ __launch_bounds__(256) void norm_kernel(const int* __restrict__ src,
                                                   const int* __restrict__ dst,
                                                   const float* __restrict__ ew,
                                                   const float* __restrict__ dis,
                                                   float* __restrict__ nrm) {
  int e = blockIdx.x * 256 + threadIdx.x;
  if (e < EE) nrm[e] = dis[src[e]] * ew[e] * dis[dst[e]];
}

// ---------------- sparse propagation: out[dst] += norm * h[src] ----------------
// One thread handles one edge x 4 consecutive features (float4 gather from L2,
// 4 global_atomic_add_f32 scatters).

template<int DIN>
__global__ __launch_bounds__(256) void prop_kernel(const float* __restrict__ h,
                                                   const int* __restrict__ src,
                                                   const int* __restrict__ dst,
                                                   const float* __restrict__ nrm,
                                                   float* __restrict__ out) {
  constexpr int G = DIN / 4;
  long long tid = (long long)blockIdx.x * 256 + threadIdx.x;
  if (tid >= (long long)EE * G) return;
  int e = (int)(tid / G);
  int g = (int)(tid % G);
  float w = nrm[e];
  const float4 v = *((const float4*)(h + (size_t)src[e] * DIN) + g);
  float* o = out + (size_t)dst[e] * DIN + (size_t)g * 4;
  atomicAdd(o + 0, w * v.x);
  atomicAdd(o + 1, w * v.y);
  atomicAdd(o + 2, w * v.z);
  atomicAdd(o + 3, w * v.w);
}

// ---------------- dense GEMM via V_WMMA_F32_16X16X4_F32 ----------------
// Block = 256 threads = 8 waves; block covers a 16-row x 128-col stripe of C.
// Wave w computes the 16x16 tile at columns [16w, 16w+16).
// Fragment layouts per CDNA5 ISA 7.12.2 (fp32, wave32):
//   A 16x4:  lane r (r=lane&15), hi=lane>>4: {A[r,k+2hi], A[r,k+2hi+1]}
//   B 4x16:  col=lane&15:        {B[k+2hi, col], B[k+2hi+1, col]}
//   C 16x16: VGPR j -> row (j + 8*hi), col = lane&15

template<int DIN, bool INIT_BIAS>
__global__ __launch_bounds__(256) void gemm_wmma(const float* __restrict__ A,
                                                 const float* __restrict__ W,   // DIN x 128
                                                 const float* __restrict__ bias,
                                                 float* __restrict__ C) {
  constexpr int DOUT = 128;
  const int lane = threadIdx.x & 31;
  const int wave = threadIdx.x >> 5;      // 0..7
  const int row0 = blockIdx.x * 16;
  const int r    = lane & 15;
  const int hi   = lane >> 4;             // 0 or 1
  const int col  = wave * 16 + r;

  v8f c;
  if (INIT_BIAS) {
    float bv = bias[col];
#pragma unroll
    for (int j = 0; j < 8; ++j) c[j] = bv;
  } else {
#pragma unroll
    for (int j = 0; j < 8; ++j)
      c[j] = C[(size_t)(row0 + j + 8 * hi) * DOUT + col];
  }

  const float* arow = A + (size_t)(row0 + r) * DIN;
#pragma unroll
  for (int k = 0; k < DIN; k += 4) {
    v2f a, b;
    a.x = arow[k + 2 * hi];
    a.y = arow[k + 2 * hi + 1];
    b.x = W[(size_t)(k + 2 * hi) * DOUT + col];
    b.y = W[(size_t)(k + 2 * hi + 1) * DOUT + col];
    c = __builtin_amdgcn_wmma_f32_16x16x4_f32(false, a, false, b, (short)0, c,
                                              false, false);
  }

#pragma unroll
  for (int j = 0; j < 8; ++j)
    C[(size_t)(row0 + j + 8 * hi) * DOUT + col] = c[j];
}

// ---------------- final layer (din=128, dout=4): scalar dot ----------------

template<bool INIT_BIAS>
__global__ __launch_bounds__(256) void gemm_final(const float* __restrict__ h,
                                                  const float* __restrict__ W,   // 128 x 4
                                                  const float* __restrict__ bias,
                                                  float* __restrict__ out) {
  int tid = blockIdx.x * 256 + threadIdx.x;
  if (tid >= NN * 4) return;
  int row = tid >> 2;
  int j   = tid & 3;
  float acc = INIT_BIAS ? bias[j] : out[tid];
  const float* hr = h + (size_t)row * 128;
#pragma unroll 8
  for (int k = 0; k < 128; ++k) acc = fmaf(hr[k], W[k * 4 + j], acc);
  out[tid] = acc;
}

// ---------------- host-side layer driver ----------------

template<int DIN>
static void run_hidden_layer(const float* h, const float* Wl, const float* bl,
                             float* out, float* pA, float* pB,
                             const int* src, const int* dst, const float* nrm,
                             hipStream_t stream) {
  const long long nh = (long long)NN * DIN;
  const int zgrid = (int)((nh + 255) / 256);
  const long long pwork = (long long)EE * (DIN / 4);
  const int pgrid = (int)((pwork + 255) / 256);
  const int ggrid = NN / 16;
  const size_t hop = (size_t)DIN * 128;

  gemm_wmma<DIN, true ><<<ggrid, 256, 0, stream>>>(h, Wl, bl, out);

  zero_kernel<<<zgrid, 256, 0, stream>>>(pA, nh);
  prop_kernel<DIN><<<pgrid, 256, 0, stream>>>(h, src, dst, nrm, pA);
  gemm_wmma<DIN, false><<<ggrid, 256, 0, stream>>>(pA, Wl + hop, bl, out);

  zero_kernel<<<zgrid, 256, 0, stream>>>(pB, nh);
  prop_kernel<DIN><<<pgrid, 256, 0, stream>>>(pA, src, dst, nrm, pB);
  gemm_wmma<DIN, false><<<ggrid, 256, 0, stream>>>(pB, Wl + 2 * hop, bl, out);

  zero_kernel<<<zgrid, 256, 0, stream>>>(pA, nh);
  prop_kernel<DIN><<<pgrid, 256, 0, stream>>>(pB, src, dst, nrm, pA);
  gemm_wmma<DIN, false><<<ggrid, 256, 0, stream>>>(pA, Wl + 3 * hop, bl, out);

  relu_kernel<<<(int)(((long long)NN * 128 + 255) / 256), 256, 0, stream>>>(
      out, (long long)NN * 128);
}

extern "C" void kernel_launch(void* const* d_in, const int* in_sizes, int n_in,
                              void* d_out, int out_size, void* d_ws, size_t ws_size,
                              hipStream_t stream) {
  (void)in_sizes; (void)n_in; (void)out_size; (void)ws_size;

  const float* x   = (const float*)d_in[0];
  const int*   src = (const int*)d_in[1];
  const int*   dst = src + EE;
  const float* ew  = (const float*)d_in[2];
  const float* W0 = (const float*)d_in[3];  const float* b0 = (const float*)d_in[4];
  const float* W1 = (const float*)d_in[5];  const float* b1 = (const float*)d_in[6];
  const float* W2 = (const float*)d_in[7];  const float* b2 = (const float*)d_in[8];
  const float* W3 = (const float*)d_in[9];  const float* b3 = (const float*)d_in[10];
  float* yout = (float*)d_out;

  // workspace layout (all offsets keep 16B alignment):
  float* dis = (float*)d_ws;          // N floats
  float* nrm = dis + NN;              // E floats
  const size_t bsz = (size_t)NN * 128;
  float* buf0 = nrm + EE;             // offset 13.2MB, 16B aligned
  float* buf1 = buf0 + bsz;
  float* buf2 = buf1 + bsz;
  float* buf3 = buf2 + bsz;

  // ---- gcn_norm ----
  zero_kernel<<<(NN + 255) / 256, 256, 0, stream>>>(dis, NN);
  deg_kernel<<<(EE + 255) / 256, 256, 0, stream>>>(dst, ew, dis);
  invsqrt_kernel<<<(NN + 255) / 256, 256, 0, stream>>>(dis);
  norm_kernel<<<(EE + 255) / 256, 256, 0, stream>>>(src, dst, ew, dis, nrm);

  // ---- layer 0: 16 -> 128 ----
  run_hidden_layer<16 >(x,    W0, b0, buf0, buf1, buf2, src, dst, nrm, stream);
  // ---- layer 1: 128 -> 128 ----
  run_hidden_layer<128>(buf0, W1, b1, buf3, buf1, buf2, src, dst, nrm, stream);
  // ---- layer 2: 128 -> 128 ----
  run_hidden_layer<128>(buf3, W2, b2, buf0, buf1, buf2, src, dst, nrm, stream);

  // ---- final layer: 128 -> 4, writes d_out ----
  {
    const float* h = buf0;
    float* pA = buf1;
    float* pB = buf2;
    const long long nh = (long long)NN * 128;
    const int zgrid = (int)((nh + 255) / 256);
    const long long pwork = (long long)EE * 32;
    const int pgrid = (int)((pwork + 255) / 256);
    const int fgrid = (NN * 4 + 255) / 256;

    gemm_final<true ><<<fgrid, 256, 0, stream>>>(h, W3, b3, yout);

    zero_kernel<<<zgrid, 256, 0, stream>>>(pA, nh);
    prop_kernel<128><<<pgrid, 256, 0, stream>>>(h, src, dst, nrm, pA);
    gemm_final<false><<<fgrid, 256, 0, stream>>>(pA, W3 + 512, b3, yout);

    zero_kernel<<<zgrid, 256, 0, stream>>>(pB, nh);
    prop_kernel<128><<<pgrid, 256, 0, stream>>>(pA, src, dst, nrm, pB);
    gemm_final<false><<<fgrid, 256, 0, stream>>>(pB, W3 + 1024, b3, yout);

    zero_kernel<<<zgrid, 256, 0, stream>>>(pA, nh);
    prop_kernel<128><<<pgrid, 256, 0, stream>>>(pB, src, dst, nrm, pA);
    gemm_final<false><<<fgrid, 256, 0, stream>>>(pA, W3 + 1536, b3, yout);
  }
}